// ContextEncoder_28810640621829
// MI455X (gfx1250) — compile-verified
//
#include <hip/hip_runtime.h>

// Segment mean-pooling (sorted segment ids) as one-hot GEMM on the CDNA5
// matrix unit: out[b*NSEG+g, :] = (1/cnt_g) * sum_{t: mask[b,t]==g} X[b,t,:]
// One wave32 per 16-segment x 16-column output tile; V_WMMA_F32_16X16X4_F32
// accumulates 4 tokens per instruction. Software-pipelined, branch-free body.

typedef __attribute__((ext_vector_type(2))) float v2f;
typedef __attribute__((ext_vector_type(8))) float v8f;

#define B_    16
#define S_    4096
#define H_    1024
#define NSEG_ 512

#define HTILES          (H_ / 16)     // 64 column tiles
#define SEG_GROUPS      (NSEG_ / 16)  // 32 segment groups per batch row
#define WAVES_PER_BLOCK 8

// Branch-free lower_bound over a sorted row of S_ = 4096 ints.
// 13 fixed iterations cover the 4097 possible answers.
__device__ __forceinline__ int lower_bound_row(const int* __restrict__ m, int key) {
    int lo = 0, hi = S_;
    #pragma unroll
    for (int i = 0; i < 13; ++i) {
        const int  mid = (lo + hi) >> 1;
        const bool lt  = (lo < hi) && (m[mid] < key);
        lo = lt ? mid + 1 : lo;
        hi = (lo < hi && !lt) ? mid : hi;
    }
    return lo;
}

__global__ __launch_bounds__(WAVES_PER_BLOCK * 32)
void seg_mean_wmma_kernel(const float* __restrict__ X,   // [B, S, H]
                          const int*   __restrict__ msk, // [B, S] sorted per row
                          float*       __restrict__ out) // [B*NSEG, H]
{
    const int wave = threadIdx.x >> 5;
    const int lane = threadIdx.x & 31;

    const int ht = blockIdx.x * WAVES_PER_BLOCK + wave;  // column tile 0..63
    const int sg = blockIdx.y;                           // segment group 0..31
    const int b  = blockIdx.z;                           // batch row

    const int* mrow  = msk + (size_t)b * S_;
    const int seg_lo = sg * 16;

    // Contiguous token range covering segments [seg_lo, seg_lo+16)
    const int start = lower_bound_row(mrow, seg_lo);
    const int end   = lower_bound_row(mrow, seg_lo + 16);

    const int m16    = lane & 15;   // A row M / B,D column N
    const int hi     = lane >> 4;   // which K half this lane holds
    const int khalf  = hi * 2;      // lanes 0-15 -> K=0,1 ; lanes 16-31 -> K=2,3
    const int target = seg_lo + m16;

    const float* xcol = X + (size_t)b * S_ * H_ + (size_t)ht * 16 + m16;

    v8f   c  = {};    // 16x16 f32 accumulator (8 VGPRs, wave32 layout)
    float fc = 0.0f;  // per-lane one-hot count accumulator

    // --- software-pipeline prologue: fetch iteration `start` ---------------
    // All indices clamped to S_-1 so loads are always valid & unconditional.
    // Sortedness guarantees tokens in [end, S) have mask >= seg_lo+16 and can
    // never match `target`; only clamped (>= S_) indices need the VALU guard.
    int ca0 = start + khalf;
    int ca1 = ca0 + 1;
    int cb0 = (ca0 < S_ - 1) ? ca0 : (S_ - 1);
    int cb1 = (ca1 < S_ - 1) ? ca1 : (S_ - 1);
    int   mv0 = mrow[cb0];
    int   mv1 = mrow[cb1];
    float xv0 = xcol[(size_t)cb0 * H_];
    float xv1 = xcol[(size_t)cb1 * H_];

    for (int t = start; t < end; t += 4) {
        // -- issue next iteration's loads (overlap with this WMMA) ----------
        const int na0 = t + 4 + khalf;
        const int na1 = na0 + 1;
        const int nb0 = (na0 < S_ - 1) ? na0 : (S_ - 1);
        const int nb1 = (na1 < S_ - 1) ? na1 : (S_ - 1);
        const int   nmv0 = mrow[nb0];
        const int   nmv1 = mrow[nb1];
        const float nxv0 = xcol[(size_t)nb0 * H_];
        const float nxv1 = xcol[(size_t)nb1 * H_];
        const int pf = (nb0 + 8 < S_) ? nb0 + 8 : (S_ - 1);
        __builtin_prefetch(&xcol[(size_t)pf * H_], 0, 1);  // global_prefetch_b8

        // -- compute with current values ------------------------------------
        v2f a;
        a.x = (mv0 == target && cb0 == ca0) ? 1.0f : 0.0f;
        a.y = (mv1 == target && cb1 == ca1) ? 1.0f : 0.0f;
        fc += a.x + a.y;

        v2f bm;
        bm.x = xv0;
        bm.y = xv1;

        // D = A(16x4 one-hot) * B(4x16 X-slice) + C
        c = __builtin_amdgcn_wmma_f32_16x16x4_f32(
                /*neg_a=*/false, a, /*neg_b=*/false, bm,
                /*c_mod=*/(short)0, c, /*reuse_a=*/false, /*reuse_b=*/false);

        // -- rotate pipeline registers --------------------------------------
        ca0 = na0; ca1 = na1; cb0 = nb0; cb1 = nb1;
        mv0 = nmv0; mv1 = nmv1; xv0 = nxv0; xv1 = nxv1;
    }

    // Token count per segment m16: this lane saw K-half `hi`; partner lane
    // (lane ^ 16) saw the other half.
    const float cnt_m16 = fc + __shfl_xor(fc, 16, 32);

    // Write the 16x16 tile: accumulator VGPR j holds row M = j + 8*hi, col N = m16.
    float* orow = out + (size_t)(b * NSEG_ + seg_lo) * H_ + (size_t)ht * 16 + m16;
    #pragma unroll
    for (int j = 0; j < 8; ++j) {
        const int   m   = j + 8 * hi;
        const float cnt = __shfl(cnt_m16, m, 32);   // cnt for segment m lives at lane m
        const float inv = 1.0f / fmaxf(cnt, 1.0f);  // empty segment -> 0
        orow[(size_t)m * H_] = c[j] * inv;
    }
}

extern "C" void kernel_launch(void* const* d_in, const int* in_sizes, int n_in,
                              void* d_out, int out_size, void* d_ws, size_t ws_size,
                              hipStream_t stream) {
    const float* X   = (const float*)d_in[0];   // hidden_states [B,S,H] f32
    const int*   msk = (const int*)d_in[1];     // output_mask   [B,S]   i32
    float*       out = (float*)d_out;           // [B*NSEG, H]   f32

    dim3 grid(HTILES / WAVES_PER_BLOCK, SEG_GROUPS, B_);  // (8, 32, 16)
    dim3 block(WAVES_PER_BLOCK * 32);                     // 256 threads = 8 wave32
    seg_mean_wmma_kernel<<<grid, block, 0, stream>>>(X, msk, out);
}